// SVDHead_87222195847751
// MI455X (gfx1250) — compile-verified
//
#include <hip/hip_runtime.h>
#include <cmath>

typedef float v2f __attribute__((ext_vector_type(2)));
typedef float v4f __attribute__((ext_vector_type(4)));
typedef float v8f __attribute__((ext_vector_type(8)));

#define SLICES  8     // slices per batch -> B*SLICES = 2048 blocks
#define NACC    22    // 22 running sums per point stream
#define PSTRIDE 33    // LDS row stride (odd -> conflict-free column writes)
#define PPB     24    // workspace stride per (batch, slice)

// ---------------------------------------------------------------------------
// Kernel 1: streaming weighted-moment accumulation (bandwidth bound).
// Each block handles one (batch, slice) chunk. Per-thread fp32 accumulation,
// then a WMMA-based ones-vector GEMM reduces the 256x22 partial matrix.
// ---------------------------------------------------------------------------
__global__ __launch_bounds__(256) void kabsch_moments_kernel(
    const float* __restrict__ src, const float* __restrict__ tgt,
    const float* __restrict__ wgt, float* __restrict__ part, int N)
{
    __shared__ float P[256 * PSTRIDE];

    const int tid   = threadIdx.x;
    const int b     = blockIdx.x / SLICES;
    const int s     = blockIdx.x - b * SLICES;
    const int chunk = N / SLICES;
    const int n0    = s * chunk;

    const size_t sb = (size_t)b * 3 * (size_t)N;
    const float* __restrict__ sx = src + sb;
    const float* __restrict__ sy = src + sb + (size_t)N;
    const float* __restrict__ sz = src + sb + 2 * (size_t)N;
    const float* __restrict__ tx = tgt + sb;
    const float* __restrict__ ty = tgt + sb + (size_t)N;
    const float* __restrict__ tz = tgt + sb + 2 * (size_t)N;
    const float* __restrict__ wp = wgt + (size_t)b * (size_t)N;

    float acc[NACC];
#pragma unroll
    for (int j = 0; j < NACC; ++j) acc[j] = 0.f;

    // Coalesced float4 nontemporal streaming (data used exactly once).
    for (int n = n0 + tid * 4; n < n0 + chunk; n += 256 * 4) {
        v4f Sx = __builtin_nontemporal_load((const v4f*)(sx + n));
        v4f Sy = __builtin_nontemporal_load((const v4f*)(sy + n));
        v4f Sz = __builtin_nontemporal_load((const v4f*)(sz + n));
        v4f Tx = __builtin_nontemporal_load((const v4f*)(tx + n));
        v4f Ty = __builtin_nontemporal_load((const v4f*)(ty + n));
        v4f Tz = __builtin_nontemporal_load((const v4f*)(tz + n));
        v4f Wv = __builtin_nontemporal_load((const v4f*)(wp + n));
#pragma unroll
        for (int e = 0; e < 4; ++e) {
            const float vsx = Sx[e], vsy = Sy[e], vsz = Sz[e];
            const float vtx = Tx[e], vty = Ty[e], vtz = Tz[e];
            const float vw  = Wv[e];
            acc[0] += vsx;  acc[1] += vsy;  acc[2] += vsz;     // sum s
            acc[3] += vtx;  acc[4] += vty;  acc[5] += vtz;     // sum t
            acc[6] += vw;                                      // sum w
            const float wsx = vw * vsx, wsy = vw * vsy, wsz = vw * vsz;
            acc[7] += wsx;  acc[8] += wsy;  acc[9] += wsz;     // sum w*s
            acc[10] += vw * vtx; acc[11] += vw * vty; acc[12] += vw * vtz; // sum w*t
            acc[13] += wsx * vtx; acc[14] += wsx * vty; acc[15] += wsx * vtz;
            acc[16] += wsy * vtx; acc[17] += wsy * vty; acc[18] += wsy * vtz;
            acc[19] += wsz * vtx; acc[20] += wsz * vty; acc[21] += wsz * vtz;
        }
    }

    // Stash per-thread partials as a 256 x 32 (padded) matrix in LDS.
#pragma unroll
    for (int j = 0; j < 32; ++j)
        P[tid * PSTRIDE + j] = (j < NACC) ? acc[j] : 0.f;
    __syncthreads();

    // Wave 0 (EXEC all-ones) reduces columns: D = ones(16x4) x P-chunk(4x16),
    // accumulated over 64 K-chunks. With A == 1 the column sums are invariant
    // to the exact K-row permutation of the B operand layout.
    if (tid < 32) {
        const int lane = tid;
        const int col  = lane & 15;
        const int krow = (lane < 16) ? 0 : 2;

        v2f a;  a[0] = 1.f; a[1] = 1.f;
        v8f c0 = {0.f, 0.f, 0.f, 0.f, 0.f, 0.f, 0.f, 0.f};
        v8f c1 = {0.f, 0.f, 0.f, 0.f, 0.f, 0.f, 0.f, 0.f};

        for (int kk = 0; kk < 64; ++kk) {
            const int p = kk * 4 + krow;
            v2f b0, b1;
            b0[0] = P[p * PSTRIDE + col];
            b0[1] = P[(p + 1) * PSTRIDE + col];
            b1[0] = P[p * PSTRIDE + 16 + col];
            b1[1] = P[(p + 1) * PSTRIDE + 16 + col];
            c0 = __builtin_amdgcn_wmma_f32_16x16x4_f32(
                     false, a, false, b0, (short)0, c0, false, false);
            c1 = __builtin_amdgcn_wmma_f32_16x16x4_f32(
                     false, a, false, b1, (short)0, c1, false, false);
        }

        // D row 0 lives in VGPR0 of lanes 0..15 (N = lane).
        if (lane < 16) {
            float* __restrict__ outp = part + ((size_t)b * SLICES + s) * PPB;
            outp[lane] = c0[0];                      // columns 0..15
            if (lane < NACC - 16)
                outp[16 + lane] = c1[0];             // columns 16..21
        }
    }
}

// ---------------------------------------------------------------------------
// Kernel 2: per-batch 3x3 Kabsch solve (trivial compute; 256 threads total).
// ---------------------------------------------------------------------------
template <int p, int q>
__device__ __forceinline__ void jrot(float A[3][3], float V[3][3])
{
    constexpr int r = 3 - p - q;
    const float apq = A[p][q];
    if (fabsf(apq) < 1e-30f) return;
    const float app = A[p][p], aqq = A[q][q];
    const float tau = (aqq - app) / (2.f * apq);
    const float t   = (tau >= 0.f ? 1.f : -1.f) / (fabsf(tau) + sqrtf(1.f + tau * tau));
    const float c   = 1.f / sqrtf(1.f + t * t);
    const float sn  = t * c;
    A[p][p] = app - t * apq;
    A[q][q] = aqq + t * apq;
    A[p][q] = 0.f; A[q][p] = 0.f;
    const float arp = A[r][p], arq = A[r][q];
    A[r][p] = c * arp - sn * arq;  A[p][r] = A[r][p];
    A[r][q] = sn * arp + c * arq;  A[q][r] = A[r][q];
#pragma unroll
    for (int k = 0; k < 3; ++k) {
        const float vkp = V[k][p], vkq = V[k][q];
        V[k][p] = c * vkp - sn * vkq;
        V[k][q] = sn * vkp + c * vkq;
    }
}

#define SWAPCOL(V, i, j)                                             \
    do {                                                             \
        float _t0 = V[0][i]; V[0][i] = V[0][j]; V[0][j] = _t0;       \
        float _t1 = V[1][i]; V[1][i] = V[1][j]; V[1][j] = _t1;       \
        float _t2 = V[2][i]; V[2][i] = V[2][j]; V[2][j] = _t2;       \
    } while (0)

__global__ __launch_bounds__(256) void kabsch_svd_kernel(
    const float* __restrict__ part, float* __restrict__ out, int B, int N)
{
    const int b = blockIdx.x * blockDim.x + threadIdx.x;
    if (b >= B) return;

    float acc[NACC];
#pragma unroll
    for (int j = 0; j < NACC; ++j) acc[j] = 0.f;
    for (int s = 0; s < SLICES; ++s) {
        const float* p = part + ((size_t)b * SLICES + s) * PPB;
#pragma unroll
        for (int j = 0; j < NACC; ++j) acc[j] += p[j];
    }

    const float invN = 1.f / (float)N;
    float sm[3] = {acc[0] * invN, acc[1] * invN, acc[2] * invN};
    float tm[3] = {acc[3] * invN, acc[4] * invN, acc[5] * invN};
    const float wsum = acc[6];
    float ws3[3] = {acc[7],  acc[8],  acc[9]};
    float wt3[3] = {acc[10], acc[11], acc[12]};

    // H[i][j] = sum(w s_i t_j) - sm_i*sum(w t_j) - tm_j*sum(w s_i) + sm_i*tm_j*sum(w)
    float H[3][3];
#pragma unroll
    for (int i = 0; i < 3; ++i)
#pragma unroll
        for (int j = 0; j < 3; ++j)
            H[i][j] = acc[13 + i * 3 + j] - sm[i] * wt3[j] - tm[j] * ws3[i]
                    + sm[i] * tm[j] * wsum;

    // A = H^T H (symmetric); Jacobi eigendecomposition -> V, lambda.
    float A[3][3];
#pragma unroll
    for (int i = 0; i < 3; ++i)
#pragma unroll
        for (int j = 0; j < 3; ++j)
            A[i][j] = H[0][i] * H[0][j] + H[1][i] * H[1][j] + H[2][i] * H[2][j];

    float V[3][3] = {{1.f, 0.f, 0.f}, {0.f, 1.f, 0.f}, {0.f, 0.f, 1.f}};
#pragma unroll
    for (int sw = 0; sw < 8; ++sw) {
        jrot<0, 1>(A, V);
        jrot<0, 2>(A, V);
        jrot<1, 2>(A, V);
    }

    float l0 = A[0][0], l1 = A[1][1], l2 = A[2][2];
    if (l0 < l1) { float t = l0; l0 = l1; l1 = t; SWAPCOL(V, 0, 1); }
    if (l1 < l2) { float t = l1; l1 = l2; l2 = t; SWAPCOL(V, 1, 2); }
    if (l0 < l1) { float t = l0; l0 = l1; l1 = t; SWAPCOL(V, 0, 1); }

    // U columns: u_j = H v_j / sigma_j (normalized); cross-product fallback.
    float U[3][3], sig[3];
#pragma unroll
    for (int j = 0; j < 3; ++j) {
        const float u0 = H[0][0] * V[0][j] + H[0][1] * V[1][j] + H[0][2] * V[2][j];
        const float u1 = H[1][0] * V[0][j] + H[1][1] * V[1][j] + H[1][2] * V[2][j];
        const float u2 = H[2][0] * V[0][j] + H[2][1] * V[1][j] + H[2][2] * V[2][j];
        const float nrm = sqrtf(u0 * u0 + u1 * u1 + u2 * u2);
        sig[j] = nrm;
        const float inv = 1.f / fmaxf(nrm, 1e-30f);
        U[0][j] = u0 * inv; U[1][j] = u1 * inv; U[2][j] = u2 * inv;
    }
    if (sig[2] < 1e-6f * fmaxf(sig[0], 1e-30f)) {
        const float cx = U[1][0] * U[2][1] - U[2][0] * U[1][1];
        const float cy = U[2][0] * U[0][1] - U[0][0] * U[2][1];
        const float cz = U[0][0] * U[1][1] - U[1][0] * U[0][1];
        const float inv = 1.f / fmaxf(sqrtf(cx * cx + cy * cy + cz * cz), 1e-30f);
        U[0][2] = cx * inv; U[1][2] = cy * inv; U[2][2] = cz * inv;
    }

    // R0 = V U^T; reflection fix via det sign on V's last column.
    float R0[3][3];
#pragma unroll
    for (int i = 0; i < 3; ++i)
#pragma unroll
        for (int j = 0; j < 3; ++j)
            R0[i][j] = V[i][0] * U[j][0] + V[i][1] * U[j][1] + V[i][2] * U[j][2];
    const float det =
          R0[0][0] * (R0[1][1] * R0[2][2] - R0[1][2] * R0[2][1])
        - R0[0][1] * (R0[1][0] * R0[2][2] - R0[1][2] * R0[2][0])
        + R0[0][2] * (R0[1][0] * R0[2][1] - R0[1][1] * R0[2][0]);
    const float f2 = (det < 0.f) ? -1.f : 1.f;

    float R[3][3];
#pragma unroll
    for (int i = 0; i < 3; ++i)
#pragma unroll
        for (int j = 0; j < 3; ++j)
            R[i][j] = V[i][0] * U[j][0] + V[i][1] * U[j][1] + f2 * V[i][2] * U[j][2];

    // t = -R sm + tm ; outputs: Rt = R^T, t_out = -(Rt t)
    float tv[3];
#pragma unroll
    for (int i = 0; i < 3; ++i)
        tv[i] = tm[i] - (R[i][0] * sm[0] + R[i][1] * sm[1] + R[i][2] * sm[2]);

#pragma unroll
    for (int i = 0; i < 3; ++i)
#pragma unroll
        for (int j = 0; j < 3; ++j)
            out[(size_t)b * 9 + i * 3 + j] = R[j][i];          // Rt

#pragma unroll
    for (int i = 0; i < 3; ++i)
        out[(size_t)B * 9 + (size_t)b * 3 + i] =
            -(R[0][i] * tv[0] + R[1][i] * tv[1] + R[2][i] * tv[2]);
}

// ---------------------------------------------------------------------------
extern "C" void kernel_launch(void* const* d_in, const int* in_sizes, int n_in,
                              void* d_out, int out_size, void* d_ws, size_t ws_size,
                              hipStream_t stream)
{
    const float* src = (const float*)d_in[0];   // (B, 3, N) fp32
    const float* tgt = (const float*)d_in[1];   // (B, 3, N) fp32
    const float* wgt = (const float*)d_in[2];   // (B, N)    fp32
    float* out = (float*)d_out;                 // B*9 (Rt) + B*3 (t)

    const int B = out_size / 12;
    const int N = in_sizes[2] / B;

    float* part = (float*)d_ws;                 // B * SLICES * PPB floats

    kabsch_moments_kernel<<<B * SLICES, 256, 0, stream>>>(src, tgt, wgt, part, N);

    const int blocks = (B + 255) / 256;
    kabsch_svd_kernel<<<blocks, 256, 0, stream>>>(part, out, B, N);
}